// HashEmbedder_14894946582565
// MI455X (gfx1250) — compile-verified
//
#include <hip/hip_runtime.h>

// out(B x 12) = W(B x 20) * T(20 x 12), done as D(16x16) = A(16x20) * T(20x16)
// per 16-point tile with five chained V_WMMA_F32_16X16X4_F32 (K padded 18->20).
//   k = 0..5   : level0 (j = k/2, e = k%2)         -> feeds cols {2j, 2j+1}
//   k = 6..17  : level1 (j = (k-6)/4, e = (k-6)%4) -> feeds cols {6+2j, 6+2j+1}
//   k = 18..19 : zero pad
// Coefficient identity: with u = floor((xc-bmin)/grid) + w, the clipped-lerp
// weight of table entry e is exactly max(0, 1 - |u - e|) (hat function), which
// also covers the bl==res clamp edge (u==res -> coef_res == 1).
// Each lane builds ONLY the 10 A-slots its lane-half feeds; the lane-half
// selection uses exact {0,1}-mask mul+fma blends so clang cannot re-merge it
// into a dynamic-index cndmask chain. The wave-invariant T operand and all
// uniform per-axis constants are hoisted out of a grid-stride tile loop.

typedef __attribute__((ext_vector_type(2))) float v2f;
typedef __attribute__((ext_vector_type(8))) float v8f;

#define WAVES_PER_BLOCK 8
#define GRID_BLOCKS 2048

// Exact two-way blend: lo_f + hi_f == 1, each in {0,1}; one product is exactly
// zero, the other passes through unrounded.
__device__ __forceinline__ float halfsel(float c_lo, float c_hi, float lo_f, float hi_f) {
  return fmaf(c_lo, lo_f, c_hi * hi_f);
}

// Hat function: max(0, 1 - |d|). |d| is a free VOP3 source modifier.
__device__ __forceinline__ float relu1(float d) {
  return fmaxf(0.0f, 1.0f - fabsf(d));
}

__global__ __launch_bounds__(256) void hashgrid_wmma_kernel(
    const float* __restrict__ x,        // (B,3)
    const float* __restrict__ box_min,  // (3,)
    const float* __restrict__ box_max,  // (3,)
    const float* __restrict__ emb0,     // (3,2,2)
    const float* __restrict__ emb1,     // (3,4,2)
    float* __restrict__ out,            // (B,12)
    int B, int ntiles)
{
  const int lane   = threadIdx.x & 31;   // wave32
  const int wave   = threadIdx.x >> 5;
  const int nwaves = gridDim.x * WAVES_PER_BLOCK;

  const bool  hi   = lane >= 16;
  const int   n    = lane & 15;          // output column (B/D lane -> N)
  const float hi_f = hi ? 1.0f : 0.0f;
  const float lo_f = 1.0f - hi_f;

  // ---- Wave-invariant T operand: this lane's column n of T(20x16), packed
  // into the five K=4 B-tiles (VGPR0 = rows 4c+0 / 4c+2, VGPR1 = 4c+1 / 4c+3).
  float bval[20];
#pragma unroll
  for (int k = 0; k < 20; ++k) {
    float v = 0.f;
    if (k < 6) {
      const int j = k >> 1, i = k & 1, nc = 2 * j;
      const float f0 = emb0[j * 4 + i * 2 + 0];
      const float f1 = emb0[j * 4 + i * 2 + 1];
      v = (n == nc) ? f0 : ((n == nc + 1) ? f1 : 0.f);
    } else if (k < 18) {
      const int kk = k - 6, j = kk >> 2, i = kk & 3, nc = 6 + 2 * j;
      const float f0 = emb1[j * 8 + i * 2 + 0];
      const float f1 = emb1[j * 8 + i * 2 + 1];
      v = (n == nc) ? f0 : ((n == nc + 1) ? f1 : 0.f);
    }
    bval[k] = v;
  }
  v2f bop[5];
#pragma unroll
  for (int c = 0; c < 5; ++c) {
    bop[c].x = halfsel(bval[4 * c + 0], bval[4 * c + 2], lo_f, hi_f);
    bop[c].y = halfsel(bval[4 * c + 1], bval[4 * c + 3], lo_f, hi_f);
  }

  // ---- Uniform per-axis constants (scalar loads; v_rcp instead of IEEE div).
  float bmn[3], bmx[3], g0[3], rg0[3], g1[3], rg1[3];
#pragma unroll
  for (int j = 0; j < 3; ++j) {
    bmn[j] = box_min[j];
    bmx[j] = box_max[j];
    g0[j]  = bmx[j] - bmn[j];              // grid, res = 1
    rg0[j] = __builtin_amdgcn_rcpf(g0[j]);
    g1[j]  = g0[j] * (1.0f / 3.0f);        // grid, res = 3
    rg1[j] = __builtin_amdgcn_rcpf(g1[j]);
  }

  // ---- Grid-stride loop over 16-point tiles (tile is wave-uniform, so EXEC
  // is all-ones at every WMMA).
  for (int tile = blockIdx.x * WAVES_PER_BLOCK + wave; tile < ntiles; tile += nwaves) {
    const int p  = tile * 16 + n;
    const int pc = min(p, B - 1);          // clamp loads for tail tile

    float U0[3], U1[3];                    // u = floor + w, per axis, per level
#pragma unroll
    for (int j = 0; j < 3; ++j) {
      const float xj = x[(size_t)pc * 3 + j];
      const float xc = fminf(fmaxf(xj, bmn[j]), bmx[j]);
      {
        const float blf  = floorf((xc - bmn[j]) * rg0[j]);
        const float vmin = fmaf(blf, g0[j], bmn[j]);
        const float den  = (vmin + g0[j]) - vmin;          // vmax - vmin
        const float w    = (xj - vmin) * __builtin_amdgcn_rcpf(den);
        U0[j] = blf + w;
      }
      {
        const float blf  = floorf((xc - bmn[j]) * rg1[j]);
        const float vmin = fmaf(blf, g1[j], bmn[j]);
        const float den  = (vmin + g1[j]) - vmin;
        const float w    = (xj - vmin) * __builtin_amdgcn_rcpf(den);
        U1[j] = blf + w;
      }
    }

    // ---- Five chained K=4 f32 WMMAs. Lane-half K mapping per chunk:
    //  c=0: lo (L0,j0) e{0,1} | hi (L0,j1) e{0,1}
    //  c=1: lo (L0,j2) e{0,1} | hi (L1,j0) e{0,1}
    //  c=2: lo (L1,j0) e{2,3} | hi (L1,j1) e{0,1}
    //  c=3: lo (L1,j1) e{2,3} | hi (L1,j2) e{0,1}
    //  c=4: lo (L1,j2) e{2,3} | hi zero pad
    v8f acc = {};
    {
      const float u = halfsel(U0[0], U0[1], lo_f, hi_f);
      v2f a; a.x = relu1(u); a.y = relu1(u - 1.0f);
      acc = __builtin_amdgcn_wmma_f32_16x16x4_f32(false, a, false, bop[0], (short)0, acc, false, false);
    }
    {
      const float u = halfsel(U0[2], U1[0], lo_f, hi_f);
      v2f a; a.x = relu1(u); a.y = relu1(u - 1.0f);
      acc = __builtin_amdgcn_wmma_f32_16x16x4_f32(false, a, false, bop[1], (short)0, acc, false, false);
    }
    {
      const float u = halfsel(U1[0], U1[1], lo_f, hi_f);
      const float d = u - 2.0f * lo_f;     // e base: 2 for lo, 0 for hi
      v2f a; a.x = relu1(d); a.y = relu1(d - 1.0f);
      acc = __builtin_amdgcn_wmma_f32_16x16x4_f32(false, a, false, bop[2], (short)0, acc, false, false);
    }
    {
      const float u = halfsel(U1[1], U1[2], lo_f, hi_f);
      const float d = u - 2.0f * lo_f;
      v2f a; a.x = relu1(d); a.y = relu1(d - 1.0f);
      acc = __builtin_amdgcn_wmma_f32_16x16x4_f32(false, a, false, bop[3], (short)0, acc, false, false);
    }
    {
      const float d = U1[2] - 2.0f;
      v2f a; a.x = relu1(d) * lo_f; a.y = relu1(d - 1.0f) * lo_f;  // hi half = pad
      acc = __builtin_amdgcn_wmma_f32_16x16x4_f32(false, a, false, bop[4], (short)0, acc, false, false);
    }

    // ---- Store: VGPR r holds row M=r (lanes 0-15) / M=r+8 (lanes 16-31),
    // col n; 12 contiguous floats per point, immediate offsets r*48B.
    if (n < 12) {
      const int base = tile * 16 + (hi ? 8 : 0);
      float* o = out + (size_t)base * 12 + n;
      if (tile * 16 + 16 <= B) {
#pragma unroll
        for (int r = 0; r < 8; ++r) o[r * 12] = acc[r];
      } else {
#pragma unroll
        for (int r = 0; r < 8; ++r)
          if (base + r < B) o[r * 12] = acc[r];
      }
    }
  }
}

extern "C" void kernel_launch(void* const* d_in, const int* in_sizes, int n_in,
                              void* d_out, int out_size, void* d_ws, size_t ws_size,
                              hipStream_t stream) {
  const float* x    = (const float*)d_in[0];
  const float* bmin = (const float*)d_in[1];
  const float* bmax = (const float*)d_in[2];
  const float* e0   = (const float*)d_in[3];
  const float* e1   = (const float*)d_in[4];
  float*       out  = (float*)d_out;

  const int B      = in_sizes[0] / 3;
  const int ntiles = (B + 15) / 16;
  int blocks = (ntiles + WAVES_PER_BLOCK - 1) / WAVES_PER_BLOCK;
  if (blocks > GRID_BLOCKS) blocks = GRID_BLOCKS;

  hipLaunchKernelGGL(hashgrid_wmma_kernel, dim3(blocks), dim3(256), 0, stream,
                     x, bmin, bmax, e0, e1, out, B, ntiles);
}